// multiheadAttentionGraphQKV_66262755442762
// MI455X (gfx1250) — compile-verified
//
#include <hip/hip_runtime.h>
#include <cstddef>

// -------- CDNA5 f32 WMMA (V_WMMA_F32_16X16X4_F32), wave32 --------
typedef __attribute__((ext_vector_type(2))) float v2f;
typedef __attribute__((ext_vector_type(8))) float v8f;

__device__ inline v8f wmma_f32_16x16x4(v2f a, v2f b, v8f c) {
  // 8 args: (neg_a, A, neg_b, B, c_mod, C, reuse_a, reuse_b)
  return __builtin_amdgcn_wmma_f32_16x16x4_f32(
      false, a, false, b, (short)0, c, false, false);
}

// -------- CDNA5 Tensor Data Mover (async tile DMA -> LDS) --------
#if defined(__has_builtin)
#  if __has_builtin(__builtin_amdgcn_tensor_load_to_lds) && \
      __has_builtin(__builtin_amdgcn_s_wait_tensorcnt)
#    define USE_TDM 1
#  endif
#endif
#ifndef USE_TDM
#  define USE_TDM 0
#endif

#if USE_TDM
typedef __attribute__((ext_vector_type(4))) unsigned int u32x4;
typedef __attribute__((ext_vector_type(4))) int i32x4;
typedef __attribute__((ext_vector_type(8))) int i32x8;

// DMA a 16x64 f32 tile (row stride 64 floats in memory) into LDS,
// inserting 1 dword of padding after every 64 dwords so the LDS image
// has row stride 65 floats (bank-conflict-free). Issue from ONE wave.
__device__ inline void tdm_load_tile16x64_padded(unsigned ldsAddr,
                                                 const float* gsrc)
{
  const unsigned long long ga = (unsigned long long)gsrc;
  // D# group 0: count=1 | lds_addr | global_addr[56:0] | type=2
  u32x4 g0 = { 1u,
               ldsAddr,
               (unsigned)ga,
               (((unsigned)(ga >> 32)) & 0x01FFFFFFu) | (2u << 30) };
  // D# group 1:
  //  s0: wg_mask=0 | data_size=2 (4B) | pad_enable | pad_interval=5 (64 dw),
  //      pad_amount=0 (1 dw)
  //  s1: tensor_dim0 = 64 (bits 79:48 low half)
  //  s2: tensor_dim1 = 384 (bits 111:80 low half)
  //  s3: tile_dim0 = 64 (bits 127:112)
  //  s4: tile_dim1 = 16 (bits 143:128), tile_dim2 = 0
  //  s5: tensor_dim0_stride = 64
  i32x8 g1 = { (int)((2u << 16) | (1u << 20) | (5u << 22)),
               (int)(64u << 16),
               (int)(384u << 16),
               (int)(64u << 16),
               16,
               64,
               0, 0 };
  i32x4 z4 = { 0, 0, 0, 0 };
#if defined(__clang_major__) && (__clang_major__ >= 23)
  i32x8 z8 = { 0, 0, 0, 0, 0, 0, 0, 0 };
  __builtin_amdgcn_tensor_load_to_lds(g0, g1, z4, z4, z8, 0);
#else
  __builtin_amdgcn_tensor_load_to_lds(g0, g1, z4, z4, 0);
#endif
}
#endif

// Generic GEMM  C[M,N] = A[M,K] @ B[K,N] + bias[N], one wave per 16x16 tile.
// A layout per lane: M = lane&15, K-pair base = 2*(lane>>4).
// B layout per lane: N = lane&15, same K-pair base.
// D layout: VGPR r -> row m0 + r + 8*(lane>>4), col n0 + (lane&15).
__global__ __launch_bounds__(256)
void gemm_bias_wmma(const float* __restrict__ A, int lda,
                    const float* __restrict__ B, int ldb,
                    const float* __restrict__ bias,
                    float* __restrict__ C, int ldc,
                    int M, int N, int K)
{
  const int wave = blockIdx.x * (blockDim.x >> 5) + (threadIdx.x >> 5);
  const int lane = threadIdx.x & 31;
  const int tilesN = N >> 4;
  const int tiles  = (M >> 4) * tilesN;
  if (wave >= tiles) return;               // wave-uniform: EXEC stays all-1s
  const int m0 = (wave / tilesN) << 4;
  const int n0 = (wave % tilesN) << 4;
  const int lm = lane & 15;
  const int khalf = (lane >> 4) << 1;      // 0 or 2
  const float* Arow = A + (size_t)(m0 + lm) * lda;
  const float* Bcol = B + (n0 + lm);
  v8f acc = {};
  for (int k = 0; k < K; k += 4) {
    const int kk = k + khalf;
    v2f a;  a.x  = Arow[kk];                a.y  = Arow[kk + 1];
    v2f bb; bb.x = Bcol[(size_t)kk * ldb];  bb.y = Bcol[(size_t)(kk + 1) * ldb];
    acc = wmma_f32_16x16x4(a, bb, acc);
  }
  const float bv = bias[n0 + lm];
  float* Crow = C + (size_t)(m0 + ((lane >> 4) << 3)) * ldc + (n0 + lm);
  #pragma unroll
  for (int r = 0; r < 8; ++r)
    Crow[(size_t)r * ldc] = acc[r] + bv;
}

// fq = q_h @ (W1+W3) + att_b[h];  gk = k_h @ (W2-W3).
// attW is (H, 192, 64): W1 rows 0..63, W2 rows 64..127, W3 rows 128..191.
// Outputs fq/gk as (H, 768, 64). One wave per 16x16 tile.
__global__ __launch_bounds__(256)
void fqgk_wmma(const float* __restrict__ q, const float* __restrict__ kp,
               const float* __restrict__ attW, const float* __restrict__ attB,
               float* __restrict__ fq, float* __restrict__ gk)
{
  int wave = blockIdx.x * (blockDim.x >> 5) + (threadIdx.x >> 5);
  const int lane = threadIdx.x & 31;
  const int nt = wave & 3;  wave >>= 2;    // 4 n-tiles (HID=64)
  const int mt = wave % 48; wave /= 48;    // 48 m-tiles (768 rows)
  const int which = wave & 1;              // 0 -> fq, 1 -> gk
  const int h = wave >> 1;
  if (h >= 8) return;                      // wave-uniform
  const int lm = lane & 15;
  const int khalf = (lane >> 4) << 1;
  const int col = nt * 16 + lm;            // e in 0..63
  const float* Arow = (which ? kp : q) + (size_t)(mt * 16 + lm) * 512 + h * 64;
  const float* Wh  = attW + (size_t)h * 192 * 64;
  const float* W12 = Wh + (which ? 64 * 64 : 0);
  const float* W3  = Wh + 128 * 64;
  const float sign = which ? -1.0f : 1.0f;
  v8f acc = {};
  for (int k = 0; k < 64; k += 4) {
    const int kk = k + khalf;
    v2f a; a.x = Arow[kk]; a.y = Arow[kk + 1];
    v2f b;
    b.x = fmaf(sign, W3[(size_t)kk * 64 + col],       W12[(size_t)kk * 64 + col]);
    b.y = fmaf(sign, W3[(size_t)(kk + 1) * 64 + col], W12[(size_t)(kk + 1) * 64 + col]);
    acc = wmma_f32_16x16x4(a, b, acc);
  }
  const float bv = which ? 0.0f : attB[h * 64 + col];
  float* D = (which ? gk : fq) + (size_t)h * 768 * 64;
  float* Drow = D + (size_t)(mt * 16 + ((lane >> 4) << 3)) * 64 + col;
  #pragma unroll
  for (int r = 0; r < 8; ++r)
    Drow[(size_t)r * 64] = acc[r] + bv;
}

// Fused: scores(i,j) = (1/8) * sum_e a[e]*leaky(fq[i,e]+gk[j,e]); softmax over j;
// msg tile (16x64) = att(16x384) @ v_hb(384x64) via WMMA. One block per (h,b,i-tile).
// gk tiles are staged into LDS by the Tensor Data Mover, double-buffered so the
// DMA of tile jt+1 overlaps the VALU score computation of tile jt.
#define N_NODES 384
__global__ __launch_bounds__(256)
void attn_softmax_msg(const float* __restrict__ fq, const float* __restrict__ gk,
                      const float* __restrict__ v, const float* __restrict__ avec,
                      float* __restrict__ msg)
{
  __shared__ float sS[16][N_NODES + 1];  // +1 pad: conflict-free WMMA A reads
  __shared__ float sFq[16][65];          // row stride 65 (TDM pad reproduces this)
  __shared__ float sGk[2][16][65];       // double buffer for async TDM
  __shared__ float sA[64];
  __shared__ float sRed[16][17];
  __shared__ float sStat[16];

  const int tid = threadIdx.x;
  const int h  = blockIdx.x / 48;
  const int b  = (blockIdx.x / 24) & 1;
  const int i0 = (blockIdx.x % 24) * 16;

  const float* fqT = fq + ((size_t)h * 768 + b * 384 + i0) * 64;
  const float* gkB = gk + ((size_t)h * 768 + b * 384) * 64;

  if (tid < 64) sA[tid] = avec[h * 64 + tid];

#if USE_TDM
  if (tid < 32) {                        // wave 0 drives the TDM (EXEC ignored)
    tdm_load_tile16x64_padded((unsigned)(size_t)&sFq[0][0],    fqT);
    tdm_load_tile16x64_padded((unsigned)(size_t)&sGk[0][0][0], gkB);
    __builtin_amdgcn_s_wait_tensorcnt(0);
  }
#else
  for (int idx = tid; idx < 16 * 64; idx += 256)
    sFq[idx >> 6][idx & 63] = fqT[idx];
#endif
  __syncthreads();

  const int ti = tid >> 4, tj = tid & 15;
  // ---- Phase A: pairwise scores (VALU; non-bilinear, not WMMA-able) ----
  for (int jt = 0; jt < 24; ++jt) {
    const int buf = jt & 1;
#if USE_TDM
    if (tid < 32) {
      if (jt + 1 < 24) {
        // prefetch next tile into the other buffer; its previous contents
        // were last read in iteration jt-1, which ended with a barrier.
        tdm_load_tile16x64_padded((unsigned)(size_t)&sGk[buf ^ 1][0][0],
                                  gkB + (size_t)(jt + 1) * 16 * 64);
        __builtin_amdgcn_s_wait_tensorcnt(1);  // tile jt complete
      } else {
        __builtin_amdgcn_s_wait_tensorcnt(0);  // last tile complete
      }
    }
#else
    for (int idx = tid; idx < 16 * 64; idx += 256)
      sGk[buf][idx >> 6][idx & 63] = gkB[(size_t)(jt * 16) * 64 + idx];
#endif
    __syncthreads();
    float s = 0.0f;
    #pragma unroll 8
    for (int e = 0; e < 64; ++e) {
      const float t  = sFq[ti][e] + sGk[buf][tj][e];
      const float lr = fmaxf(t, 0.0f) + 0.2f * fminf(t, 0.0f);  // leaky_relu
      s = fmaf(sA[e], lr, s);
    }
    sS[ti][jt * 16 + tj] = s * 0.125f;   // 1/sqrt(64)
    __syncthreads();
  }

  // ---- Phase B: row softmax over 384 cols (16 threads per row) ----
  {
    const int r = ti, c = tj;
    float m = -3.0e38f;
    for (int j = c; j < N_NODES; j += 16) m = fmaxf(m, sS[r][j]);
    sRed[r][c] = m;
    __syncthreads();
    if (c == 0) {
      float mm = sRed[r][0];
      #pragma unroll
      for (int x = 1; x < 16; ++x) mm = fmaxf(mm, sRed[r][x]);
      sStat[r] = mm;
    }
    __syncthreads();
    const float rowm = sStat[r];
    float acc = 0.0f;
    for (int j = c; j < N_NODES; j += 16) {
      const float e = __expf(sS[r][j] - rowm);
      sS[r][j] = e;
      acc += e;
    }
    sRed[r][c] = acc;
    __syncthreads();
    if (c == 0) {
      float t = 0.0f;
      #pragma unroll
      for (int x = 0; x < 16; ++x) t += sRed[r][x];
      sStat[r] = 1.0f / t;
    }
    __syncthreads();
    const float rinv = sStat[r];
    for (int j = c; j < N_NODES; j += 16) sS[r][j] *= rinv;
    __syncthreads();
  }

  // ---- Phase C: msg tile = att @ v via f32 WMMA (waves 0..3, one n-tile each) ----
  const int wave = tid >> 5, lane = tid & 31;
  if (wave < 4) {                          // wave-uniform branch
    const int n0 = wave * 16;
    const int lm = lane & 15;
    const int khalf = (lane >> 4) << 1;
    const float* vh = v + (size_t)(b * 384) * 512 + h * 64 + n0 + lm;
    v8f acc = {};
    for (int k = 0; k < N_NODES; k += 4) {
      const int kk = k + khalf;
      v2f a;  a.x  = sS[lm][kk];             a.y  = sS[lm][kk + 1];
      v2f bb; bb.x = vh[(size_t)kk * 512];   bb.y = vh[(size_t)(kk + 1) * 512];
      acc = wmma_f32_16x16x4(a, bb, acc);
    }
    float* mrow = msg + (size_t)(b * 384 + i0 + ((lane >> 4) << 3)) * 512
                      + h * 64 + n0 + lm;
    #pragma unroll
    for (int r = 0; r < 8; ++r)
      mrow[(size_t)r * 512] = acc[r];
  }
}

extern "C" void kernel_launch(void* const* d_in, const int* in_sizes, int n_in,
                              void* d_out, int out_size, void* d_ws, size_t ws_size,
                              hipStream_t stream)
{
  const float* nodes = (const float*)d_in[0];
  const float* WQ   = (const float*)d_in[1];
  const float* bQ   = (const float*)d_in[2];
  const float* WK   = (const float*)d_in[3];
  const float* bK   = (const float*)d_in[4];
  const float* WV   = (const float*)d_in[5];
  const float* bV   = (const float*)d_in[6];
  const float* attW = (const float*)d_in[7];
  const float* attB = (const float*)d_in[8];
  const float* avec = (const float*)d_in[9];
  const float* Wout = (const float*)d_in[10];
  const float* bout = (const float*)d_in[11];
  float* out = (float*)d_out;
  float* ws  = (float*)d_ws;

  const size_t SZ = 768 * 512;             // one (B*N, 512) matrix
  float* q   = ws + 0 * SZ;
  float* kb  = ws + 1 * SZ;
  float* vb  = ws + 2 * SZ;
  float* fq  = ws + 3 * SZ;                // (H, 768, 64)
  float* gk  = ws + 4 * SZ;                // (H, 768, 64)
  float* msg = ws + 5 * SZ;                // (B*N, H*VD)

  dim3 blk(256);
  // QKV projections: 768x512 @ 512x512, 1536 tiles, 8 waves/block -> 192 blocks
  gemm_bias_wmma<<<192, blk, 0, stream>>>(nodes, 512, WQ, 512, bQ, q,  512, 768, 512, 512);
  gemm_bias_wmma<<<192, blk, 0, stream>>>(nodes, 512, WK, 512, bK, kb, 512, 768, 512, 512);
  gemm_bias_wmma<<<192, blk, 0, stream>>>(nodes, 512, WV, 512, bV, vb, 512, 768, 512, 512);
  // fq/gk: 8 heads x 2 mats x 48 x 4 tiles = 3072 waves -> 384 blocks
  fqgk_wmma<<<384, blk, 0, stream>>>(q, kb, attW, attB, fq, gk);
  // fused scores + softmax + att@v: H*B*24 = 384 blocks
  attn_softmax_msg<<<384, blk, 0, stream>>>(fq, gk, vb, avec, msg);
  // output projection
  gemm_bias_wmma<<<192, blk, 0, stream>>>(msg, 512, Wout, 512, bout, out, 512, 768, 512, 512);
}